// PoseConsistencyLoss_68676527063609
// MI455X (gfx1250) — compile-verified
//
#include <hip/hip_runtime.h>
#include <math.h>

typedef __attribute__((ext_vector_type(2))) float v2f;
typedef __attribute__((ext_vector_type(8))) float v8f;

#define N_SPLATS 65536
#define N_LM     2048
#define M_TILES  (N_LM / 16)          // 128 landmark tiles
#define NCHUNK   8                    // splat chunks (grid.y)
#define CHUNK    (N_SPLATS / NCHUNK)  // 8192 splats per chunk
#define WAVES_PB 8                    // 256-thread block = 8 wave32
#define ROWS_PW  (CHUNK / WAVES_PB)   // 1024 splats per wave
#define TILES_PW (ROWS_PW / 16)       // 64 WMMA tiles per wave
#define MAX_DIST 1.0f
#define NLANES_CHAIN 4                // independent argmin chains per lane

// ---- workspace layout (float indices) ----
#define WS_LCAM  0                        // [N_LM*3] landmarks in camera frame
#define WS_BPREP (WS_LCAM + N_LM * 3)     // [N_LM*4] (-2bx,-2by,-2bz,|b|^2)
#define WS_PMIN  (WS_BPREP + N_LM * 4)    // [N_LM*NCHUNK] partial min d^2
#define WS_PIDX  (WS_PMIN + N_LM * NCHUNK)// [N_LM*NCHUNK] partial argmin (int)
#define WS_ACC   (WS_PIDX + N_LM * NCHUNK)// [4] accumulators

// Stage 0: transform landmarks into camera frame, prep WMMA B operand, zero accs.
__global__ void pcl_prep(const float* __restrict__ lm3d,
                         const float* __restrict__ pose,
                         float* __restrict__ ws) {
  int m = blockIdx.x * blockDim.x + threadIdx.x;
  if (m < 4) ws[WS_ACC + m] = 0.0f;
  if (m >= N_LM) return;
  float x = lm3d[3 * m + 0], y = lm3d[3 * m + 1], z = lm3d[3 * m + 2];
  float cx = pose[0] * x + pose[1] * y + pose[2]  * z + pose[3];
  float cy = pose[4] * x + pose[5] * y + pose[6]  * z + pose[7];
  float cz = pose[8] * x + pose[9] * y + pose[10] * z + pose[11];
  ws[WS_LCAM + 3 * m + 0] = cx;
  ws[WS_LCAM + 3 * m + 1] = cy;
  ws[WS_LCAM + 3 * m + 2] = cz;
  ws[WS_BPREP + 4 * m + 0] = -2.0f * cx;
  ws[WS_BPREP + 4 * m + 1] = -2.0f * cy;
  ws[WS_BPREP + 4 * m + 2] = -2.0f * cz;
  ws[WS_BPREP + 4 * m + 3] = cx * cx + cy * cy + cz * cz;  // |b|^2
}

// Stage 1: WMMA distance tiles + per-column min/argmin over this chunk.
// grid = (M_TILES, NCHUNK), block = 256 (8 wave32).
__global__ __launch_bounds__(256) void pcl_dist_argmin(
    const float* __restrict__ splats, float* __restrict__ ws) {
  __shared__ float smin[WAVES_PB][16];
  __shared__ int   sidx[WAVES_PB][16];

  const int lane = threadIdx.x & 31;
  const int wave = threadIdx.x >> 5;
  const int col  = lane & 15;          // landmark column owned by this lane
  const bool hi  = lane >= 16;         // upper half-wave holds K=2,3 / M=+8
  const int lm   = blockIdx.x * 16 + col;

  // B operand: 4x16, col striped across lanes; VGPR0=K0/K2, VGPR1=K1/K3.
  const float* bp = ws + WS_BPREP + 4 * lm;
  v2f B;
  B.x = hi ? bp[2] : bp[0];            // K2 = -2bz : K0 = -2bx
  B.y = hi ? 1.0f  : bp[1];            // K3 = 1    : K1 = -2by
  const float bn2 = bp[3];             // |b|^2, added per column after WMMA

  // 4 independent argmin chains -> 4x shorter carried VALU dependency.
  float bestd[NLANES_CHAIN];
  int   besti[NLANES_CHAIN];
#pragma unroll
  for (int k = 0; k < NLANES_CHAIN; ++k) { bestd[k] = 3.4e38f; besti[k] = 0; }

  const int rowbase = blockIdx.y * CHUNK + wave * ROWS_PW;

  for (int t = 0; t < TILES_PW; ++t) {
    const int r = rowbase + t * 16 + col;      // both halves load same row set
    const float* sp = splats + 3 * r;
    __builtin_prefetch(sp + 3 * 16 * 4, 0, 3); // WGP-scope prefetch, 4 tiles ahead
    float sx = sp[0], sy = sp[1], sz = sp[2];
    float an2 = sx * sx + sy * sy + sz * sz;
    v2f A;
    A.x = hi ? sz  : sx;                       // K2 : K0
    A.y = hi ? an2 : sy;                       // K3 = |a|^2 : K1
    v8f c = {};
    // D = A(16x4) x B(4x16) = |a|^2 - 2 a.b  (per row/col pair)
    c = __builtin_amdgcn_wmma_f32_16x16x4_f32(
        /*neg_a=*/false, A, /*neg_b=*/false, B,
        /*c_mod=*/(short)0, c, /*reuse_a=*/false, /*reuse_b=*/false);
#pragma unroll
    for (int v = 0; v < 8; ++v) {
      float d2  = c[v] + bn2;                  // full squared distance
      int   row = rowbase + t * 16 + v + (hi ? 8 : 0);
      int   k   = v & (NLANES_CHAIN - 1);      // distribute over chains
      bool better = d2 < bestd[k];             // branchless: keeps EXEC intact
      bestd[k] = better ? d2  : bestd[k];
      besti[k] = better ? row : besti[k];
    }
  }

  // Merge chains (rows in a chain are interleaved, so prefer lower index on tie).
  float bd0 = bestd[0]; int bi0 = besti[0];
#pragma unroll
  for (int k = 1; k < NLANES_CHAIN; ++k) {
    bool better = (bestd[k] < bd0) || (bestd[k] == bd0 && besti[k] < bi0);
    bd0 = better ? bestd[k] : bd0;
    bi0 = better ? besti[k] : bi0;
  }

  // Merge the two half-lane views of the same column (lane <-> lane^16).
  float od = __shfl_xor(bd0, 16, 32);
  int   oi = __shfl_xor(bi0, 16, 32);
  bool better = (od < bd0) || (od == bd0 && oi < bi0);
  bd0 = better ? od : bd0;
  bi0 = better ? oi : bi0;

  if (!hi) { smin[wave][col] = bd0; sidx[wave][col] = bi0; }
  __syncthreads();

  if (wave == 0 && !hi) {
    float bd = smin[0][col]; int bi = sidx[0][col];
#pragma unroll
    for (int w = 1; w < WAVES_PB; ++w) {
      float d = smin[w][col]; int i = sidx[w][col];
      bool b2 = (d < bd) || (d == bd && i < bi);
      bd = b2 ? d : bd; bi = b2 ? i : bi;
    }
    ws[WS_PMIN + lm * NCHUNK + blockIdx.y] = bd;
    ((int*)ws)[WS_PIDX + lm * NCHUNK + blockIdx.y] = bi;
  }
}

// Stage 2: per-landmark final reduce + scalar accumulation.
__global__ void pcl_reduce(const float* __restrict__ splats,
                           float* __restrict__ ws) {
  int m = blockIdx.x * blockDim.x + threadIdx.x;
  if (m >= N_LM) return;
  const float* pm = ws + WS_PMIN + m * NCHUNK;
  const int*   pi = (const int*)ws + WS_PIDX + m * NCHUNK;
  float bd = pm[0]; int bi = pi[0];
#pragma unroll
  for (int c = 1; c < NCHUNK; ++c) {          // chunks ascend in splat index
    float d = pm[c]; int i = pi[c];
    bool better = (d < bd) || (d == bd && i < bi);
    bd = better ? d : bd; bi = better ? i : bi;
  }
  float mind = sqrtf(fmaxf(bd, 0.0f));
  if (mind < MAX_DIST) {
    const float* lc = ws + WS_LCAM + 3 * m;
    const float* sp = splats + 3 * bi;
    float dx = sp[0] - lc[0], dy = sp[1] - lc[1], dz = sp[2] - lc[2];
    float* acc = ws + WS_ACC;
    atomicAdd(&acc[0], dx * dx + dy * dy + dz * dz);  // sqerr numerator
    atomicAdd(&acc[1], 1.0f);                         // num_matches
    atomicAdd(&acc[2], mind);                         // sum of min distances
  }
}

// Stage 3: finalize the three scalars.
__global__ void pcl_finalize(const float* __restrict__ acc,
                             float* __restrict__ out) {
  float se = acc[0], nm = acc[1], sd = acc[2];
  out[0] = (nm > 0.0f) ? (se / fmaxf(nm * 3.0f, 1.0f)) : 0.0f; // LANDMARK_WEIGHT=1
  out[1] = nm;                                                  // num_matches
  out[2] = (nm > 0.0f) ? (sd / fmaxf(nm, 1.0f)) : 0.0f;        // mean_distance
}

extern "C" void kernel_launch(void* const* d_in, const int* in_sizes, int n_in,
                              void* d_out, int out_size, void* d_ws, size_t ws_size,
                              hipStream_t stream) {
  const float* splats = (const float*)d_in[0];  // [65536,3]
  const float* pose   = (const float*)d_in[1];  // [4,4]
  const float* lm3d   = (const float*)d_in[2];  // [2048,3]
  // d_in[3] landmarks_2d and d_in[4] camera_intrinsics are unused by the reference.
  float* ws  = (float*)d_ws;
  float* out = (float*)d_out;

  pcl_prep<<<dim3(N_LM / 256), 256, 0, stream>>>(lm3d, pose, ws);
  pcl_dist_argmin<<<dim3(M_TILES, NCHUNK), 256, 0, stream>>>(splats, ws);
  pcl_reduce<<<dim3(N_LM / 256), 256, 0, stream>>>(splats, ws);
  pcl_finalize<<<1, 1, 0, stream>>>(ws + WS_ACC, out);
}